// GELU103_70428873720422
// MI455X (gfx1250) — compile-verified
//
#include <hip/hip_runtime.h>
#include <math.h>

// ---------------------------------------------------------------------------
// GELU + EMA-stat surprise gate, MI455X (gfx1250).
// Bandwidth-bound: 268 MB traffic -> ~11.5us floor @ 23.3 TB/s.
// Single pass over x: TDM async row load to LDS, per-thread register compute,
// WMMA-based block reductions, wave-local radix-select top-K, NT stores.
// ---------------------------------------------------------------------------

#define DCH   2048
#define KTOP  256
#define TPB   256

typedef float    v2f  __attribute__((ext_vector_type(2)));
typedef float    v4f  __attribute__((ext_vector_type(4)));
typedef float    v8f  __attribute__((ext_vector_type(8)));
typedef unsigned v4u  __attribute__((ext_vector_type(4)));
typedef int      v4i  __attribute__((ext_vector_type(4)));
typedef int      v8i  __attribute__((ext_vector_type(8)));

#if __has_builtin(__builtin_amdgcn_tensor_load_to_lds) && __has_builtin(__builtin_amdgcn_s_wait_tensorcnt)
#define HAVE_TDM 1
#else
#define HAVE_TDM 0
#endif

// ---------------------------------------------------------------------------
// Sum 2x256 partials using v_wmma_f32_16x16x4_f32 against an all-ones B.
// With B = ones, D[m][n] = rowsum_A[m] broadcast over n; summing each lane's
// 8 accumulator regs and adding the lane^16 partner totals all 64 A entries,
// so 4 accumulating WMMAs cover all 256 partials exactly once.
// ---------------------------------------------------------------------------
__device__ __forceinline__ void block_reduce2_wmma(float a, float b,
                                                   float* red, float* res) {
  const int t = threadIdx.x;
  red[t] = a;
  red[TPB + t] = b;
  __syncthreads();
  if (t < 32) {
    v2f ones; ones[0] = 1.0f; ones[1] = 1.0f;
    v8f c0 = {};
    v8f c1 = {};
#pragma unroll
    for (int i = 0; i < 4; ++i) {
      v2f A; A[0] = red[i * 64 + 2 * t]; A[1] = red[i * 64 + 2 * t + 1];
      c0 = __builtin_amdgcn_wmma_f32_16x16x4_f32(false, A, false, ones,
                                                 (short)0, c0, false, false);
      v2f B; B[0] = red[TPB + i * 64 + 2 * t]; B[1] = red[TPB + i * 64 + 2 * t + 1];
      c1 = __builtin_amdgcn_wmma_f32_16x16x4_f32(false, B, false, ones,
                                                 (short)0, c1, false, false);
    }
    float s0 = c0[0] + c0[1] + c0[2] + c0[3] + c0[4] + c0[5] + c0[6] + c0[7];
    float s1 = c1[0] + c1[1] + c1[2] + c1[3] + c1[4] + c1[5] + c1[6] + c1[7];
    s0 += __shfl_xor(s0, 16, 32);
    s1 += __shfl_xor(s1, 16, 32);
    if (t == 0) { res[0] = s0; res[1] = s1; }
  }
  __syncthreads();
}

// ---------------------------------------------------------------------------
// Kernel 1: per-channel inv_std, unit ema_u, and the three gate scalars.
// ws layout: [0,D) inv_std | [D,2D) ema_u | [2D..] tau, sigma, w
// ---------------------------------------------------------------------------
__global__ void __launch_bounds__(TPB) prep_kernel(
    const float* __restrict__ ema_mean, const float* __restrict__ ema_sq,
    const float* __restrict__ ema_out, const float* __restrict__ log_tau,
    const float* __restrict__ log_sigma_raw, const float* __restrict__ log_w_raw,
    float* __restrict__ ws) {
  __shared__ float sred[TPB];
  const int t = threadIdx.x;
  float ss = 0.0f;
  for (int i = t; i < DCH; i += TPB) {
    float m = ema_mean[i];
    float v = fmaxf(ema_sq[i] - m * m, 1e-6f);
    ws[i] = 1.0f / sqrtf(v);
    float eo = ema_out[i];
    ss += eo * eo;
  }
  sred[t] = ss;
  __syncthreads();
  for (int off = TPB / 2; off > 0; off >>= 1) {
    if (t < off) sred[t] += sred[t + off];
    __syncthreads();
  }
  const float inv_n = 1.0f / (sqrtf(sred[0]) + 1e-8f);
  for (int i = t; i < DCH; i += TPB) ws[DCH + i] = ema_out[i] * inv_n;
  if (t == 0) {
    ws[2 * DCH + 0] = expf(log_tau[0]);
    ws[2 * DCH + 1] = log1pf(expf(log_sigma_raw[0])) + 0.01f;  // softplus + .01
    ws[2 * DCH + 2] = log1pf(expf(log_w_raw[0]));              // softplus
  }
}

// ---------------------------------------------------------------------------
// Kernel 2: one row (2048 ch) per block. TDM async load -> LDS, register
// compute, WMMA reductions, 4-level radix select with wave-local suffix scan
// (3 barriers per level instead of ~26).
// ---------------------------------------------------------------------------
__global__ void __launch_bounds__(TPB) gate_kernel(
    const float* __restrict__ x, const float* __restrict__ ema_mean,
    const float* __restrict__ ws, float* __restrict__ out) {
  __shared__ __align__(16) float lds_x[DCH];   // 8 KB row buffer
  __shared__ float red[2 * TPB];
  __shared__ unsigned hist[TPB];
  __shared__ unsigned sel[2];                  // selected bin, remaining rank
  __shared__ float rowres[4];                  // sumsq, dot, sum_gt, cnt_gt

  const int t = threadIdx.x;
  const int row = blockIdx.x;
  const float* xrow = x + (size_t)row * DCH;

  // ---- async row fetch: Tensor Data Mover, wave 0 only -------------------
#if HAVE_TDM
  if (t < 32) {
    const unsigned lds_addr = (unsigned)(unsigned long long)(&lds_x[0]);
    const unsigned long long ga = (unsigned long long)(const void*)xrow;
    v4u g0;
    g0[0] = 1u;                                        // count = 1 descriptor
    g0[1] = lds_addr;                                  // lds_addr (bytes)
    g0[2] = (unsigned)(ga & 0xFFFFFFFFu);              // global_addr[31:0]
    g0[3] = (unsigned)((ga >> 32) & 0x01FFFFFFu)       // global_addr[56:32]
            | (2u << 30);                              // type = 2 (image)
    v8i g1;
    g1[0] = (int)(2u << 16);                           // data_size = 4 bytes
    g1[1] = (int)((unsigned)DCH << 16);                // tensor_dim0 = 2048
    g1[2] = (int)(1u << 16);                           // tensor_dim1 = 1
    g1[3] = (int)((unsigned)DCH << 16);                // tile_dim0 = 2048
    g1[4] = 1;                                         // tile_dim1 = 1
    g1[5] = DCH;                                       // tensor_dim0_stride
    g1[6] = 0;
    g1[7] = 0;
    v4i gz4; gz4[0] = 0; gz4[1] = 0; gz4[2] = 0; gz4[3] = 0;
    v8i gz8; gz8[0] = 0; gz8[1] = 0; gz8[2] = 0; gz8[3] = 0;
    gz8[4] = 0; gz8[5] = 0; gz8[6] = 0; gz8[7] = 0;
    __builtin_amdgcn_tensor_load_to_lds(g0, g1, gz4, gz4, gz8, 0);
    __builtin_amdgcn_s_wait_tensorcnt((short)0);
  }
#else
  for (int i = t; i < DCH / 4; i += TPB)
    ((v4f*)lds_x)[i] = ((const v4f*)xrow)[i];
#endif
  __syncthreads();

  // ---- per-thread: 8 elements in registers -------------------------------
  const float* inv_std = ws;
  const float* ema_u   = ws + DCH;
  const float tau   = ws[2 * DCH + 0];
  const float sigma = ws[2 * DCH + 1];
  const float w     = ws[2 * DCH + 2];

  float    outv[8];
  unsigned zb[8];
  float sq = 0.0f, dp = 0.0f;
#pragma unroll
  for (int h = 0; h < 2; ++h) {
    v4f xv = ((const v4f*)lds_x)[2 * t + h];
    v4f mv = ((const v4f*)ema_mean)[2 * t + h];
    v4f iv = ((const v4f*)inv_std)[2 * t + h];
    v4f uv = ((const v4f*)ema_u)[2 * t + h];
#pragma unroll
    for (int j = 0; j < 4; ++j) {
      const float xx = xv[j];
      const float g = 0.5f * xx * (1.0f + erff(xx * 0.70710678118654752f));
      outv[h * 4 + j] = g;
      sq += g * g;
      dp += g * uv[j];
      const float z = (xx - mv[j]) * iv[j];
      zb[h * 4 + j] = __float_as_uint(fabsf(z));
    }
  }
  block_reduce2_wmma(sq, dp, red, rowres);   // rowres[0]=||out||^2, [1]=dot

  // ---- exact top-K threshold: 4-level radix select on |z| bit patterns.
  // 256-bin histogram in LDS; suffix scan + bin selection done entirely in
  // wave 0 (8 register bins/lane + 5-step shfl wave scan): 3 barriers/level.
  unsigned prefix = 0;
  for (int lvl = 3; lvl >= 0; --lvl) {
    hist[t] = 0;
    __syncthreads();
    const int sh = lvl * 8;
#pragma unroll
    for (int i = 0; i < 8; ++i) {
      const unsigned b = zb[i];
      bool ok = (lvl == 3);
      if (!ok) ok = ((b >> (sh + 8)) == prefix);
      if (ok) atomicAdd(&hist[(b >> sh) & 0xFFu], 1u);
    }
    __syncthreads();
    if (t < 32) {
      unsigned h[8];
      unsigned tot = 0;
#pragma unroll
      for (int j = 0; j < 8; ++j) { h[j] = hist[t * 8 + j]; tot += h[j]; }
      // inclusive suffix scan of lane totals: S = sum over lanes [t..31]
      unsigned S = tot;
#pragma unroll
      for (int off = 1; off < 32; off <<= 1) {
        const unsigned v = __shfl_down(S, off, 32);
        if (t + off < 32) S += v;
      }
      const unsigned above = S - tot;                 // bins in higher lanes
      const unsigned want = (lvl == 3) ? (unsigned)KTOP : sel[1];
      unsigned run = above;                           // suf[8t + j + 1]
#pragma unroll
      for (int j = 7; j >= 0; --j) {
        const unsigned nxt = run;
        run += h[j];                                  // suf[8t + j]
        if (run >= want && nxt < want) {
          sel[0] = (unsigned)(t * 8 + j);
          sel[1] = want - nxt;
        }
      }
    }
    __syncthreads();
    prefix = (prefix << 8) | sel[0];
  }

  // ---- top-K sum: strictly-greater + ties at the threshold ---------------
  const unsigned thr_bits = prefix;
  const float thr = __uint_as_float(thr_bits);
  float sgt = 0.0f, cgt = 0.0f;
#pragma unroll
  for (int i = 0; i < 8; ++i) {
    if (zb[i] > thr_bits) { sgt += __uint_as_float(zb[i]); cgt += 1.0f; }
  }
  block_reduce2_wmma(sgt, cgt, red, rowres + 2);

  // ---- gate and streamed write ------------------------------------------
  const float topk_mean = (rowres[2] + ((float)KTOP - rowres[3]) * thr)
                          * (1.0f / (float)KTOP);
  const float cos_out = rowres[1] / fmaxf(sqrtf(rowres[0]), 1e-12f);
  const float gate = expf(-tau * cos_out) * (1.0f + w * tanhf(sigma * topk_mean));

  float* orow = out + (size_t)row * DCH;
  v4f o0, o1;
#pragma unroll
  for (int j = 0; j < 4; ++j) {
    o0[j] = outv[j] * gate;
    o1[j] = outv[4 + j] * gate;
  }
  __builtin_nontemporal_store(o0, (v4f*)orow + 2 * t);
  __builtin_nontemporal_store(o1, (v4f*)orow + 2 * t + 1);
}

// ---------------------------------------------------------------------------
extern "C" void kernel_launch(void* const* d_in, const int* in_sizes, int n_in,
                              void* d_out, int out_size, void* d_ws, size_t ws_size,
                              hipStream_t stream) {
  const float* x             = (const float*)d_in[0];
  const float* ema_mean      = (const float*)d_in[1];
  const float* ema_sq        = (const float*)d_in[2];
  const float* ema_out       = (const float*)d_in[3];
  const float* log_tau       = (const float*)d_in[4];
  const float* log_sigma_raw = (const float*)d_in[5];
  const float* log_w_raw     = (const float*)d_in[6];
  float* ws = (float*)d_ws;
  const int n_rows = in_sizes[0] / DCH;   // B*T = 16384

  prep_kernel<<<1, TPB, 0, stream>>>(ema_mean, ema_sq, ema_out, log_tau,
                                     log_sigma_raw, log_w_raw, ws);
  gate_kernel<<<n_rows, TPB, 0, stream>>>(x, ema_mean, ws, (float*)d_out);
}